// SpanInformationCollectingLayer_11536282157203
// MI455X (gfx1250) — compile-verified
//
#include <hip/hip_runtime.h>

// ---------------------------------------------------------------------------
// SpanInformationCollectingLayer for MI455X (gfx1250, wave32, WMMA)
//
// out[b,(i,j),d] = tanh( W1·hi + W2·hj + W3·(hi-hj) + W4·(hi*hj) + bias )
//                = tanh( U[b,i,d] + V[b,j,d] + P[b,(i,j),d] + bias )
//   U = (W1+W3)·h  , V = (W2-W3)·h   (precomputed, 2MB workspace)
//   P = W4·(hi*hj)  -> the only true per-pair GEMM (8.66 GFLOP vs 34.6 naive)
// All matmuls via V_WMMA_F32_16X16X4_F32 (f32 in / f32 acc).
// ---------------------------------------------------------------------------

#define Nn   128
#define Dd   256
#define Sp   8256          // N*(N+1)/2, == 129 * 64 exactly
#define TSp  64            // pair-tile per block
#define TDd  64            // d-tile per block
#define STRH 260           // padded LDS row stride (floats), mult of 4, bank-skewed
#define STRW 260

typedef __attribute__((ext_vector_type(2))) float v2f;
typedef __attribute__((ext_vector_type(8))) float v8f;

__device__ __forceinline__ v8f wmma_f32(v2f a, v2f b, v8f c) {
  // V_WMMA_F32_16X16X4_F32 : 16x16 f32 tile += A(16x4) * B(4x16)
  return __builtin_amdgcn_wmma_f32_16x16x4_f32(
      /*neg_a=*/false, a, /*neg_b=*/false, b,
      /*c_mod=*/(short)0, c, /*reuse_a=*/false, /*reuse_b=*/false);
}

__device__ __forceinline__ float fast_tanh(float x) {
  // tanh(x) = sign(x) * (1 - 2/(exp2(2|x|*log2e) + 1)); exact limits at +-inf
  float ax = __builtin_fabsf(x);
  float q  = __builtin_amdgcn_exp2f(ax * 2.8853900817779268f);
  float t  = 1.0f - 2.0f * __builtin_amdgcn_rcpf(q + 1.0f);
  return __builtin_copysignf(t, x);
}

__device__ __forceinline__ int rowStart(int i) {
  // first flat pair index of row i in triu(N)
  return i * Nn - ((i * (i - 1)) >> 1);
}

// ---------------------------------------------------------------------------
// Kernel 1: U = h @ (W1+W3)^T, V = h @ (W2-W3)^T over flattened rows (B*N, D)
// Grid: (16 row-tiles of 64, 4 d-tiles of 64), block 256 (8 wave32).
// ---------------------------------------------------------------------------
__global__ __launch_bounds__(256, 1)
void uv_precompute_kernel(const float* __restrict__ h,
                          const float* __restrict__ W,
                          float* __restrict__ U, float* __restrict__ V) {
  extern __shared__ float smem[];
  float* lds_a   = smem;                    // 64 x STRH : rows of h
  float* lds_w13 = lds_a   + 64 * STRH;     // 64 x STRW : (W1+W3)[d][k]
  float* lds_w23 = lds_w13 + 64 * STRW;     // 64 x STRW : (W2-W3)[d][k]

  const int r0  = blockIdx.x * 64;          // flattened b*N+i row base
  const int d0  = blockIdx.y * TDd;
  const int tid = threadIdx.x;

  const float4* hg = (const float4*)h + (size_t)r0 * (Dd / 4);
  for (int idx = tid; idx < 64 * (Dd / 4); idx += 256) {
    int row = idx >> 6, c4 = idx & 63;
    float4 v = hg[row * (Dd / 4) + c4];
    *(float4*)&lds_a[row * STRH + c4 * 4] = v;
  }
  const float4* wg = (const float4*)W;
  for (int idx = tid; idx < 64 * (Dd / 4); idx += 256) {
    int row = idx >> 6, c4 = idx & 63;
    const float4* wrow = wg + (size_t)(d0 + row) * 256;   // 1024 f = 256 float4 per row
    float4 wa = wrow[c4];          // W1[d][k]
    float4 wb = wrow[64 + c4];     // W2[d][k]
    float4 wc = wrow[128 + c4];    // W3[d][k]
    float4 s13 = make_float4(wa.x + wc.x, wa.y + wc.y, wa.z + wc.z, wa.w + wc.w);
    float4 s23 = make_float4(wb.x - wc.x, wb.y - wc.y, wb.z - wc.z, wb.w - wc.w);
    *(float4*)&lds_w13[row * STRW + c4 * 4] = s13;
    *(float4*)&lds_w23[row * STRW + c4 * 4] = s23;
  }
  __syncthreads();

  const int l = tid & 31, wv = tid >> 5;
  const int lmod = l & 15, lhi = l >> 4;
  const int m0 = (wv & 3) * 16, n0 = (wv >> 2) * 32;

  const float* aptr = lds_a   + (m0 + lmod) * STRH + 2 * lhi;
  const float* b13a = lds_w13 + (n0 + lmod) * STRW + 2 * lhi;
  const float* b13b = b13a + 16 * STRW;
  const float* b23a = lds_w23 + (n0 + lmod) * STRW + 2 * lhi;
  const float* b23b = b23a + 16 * STRW;

  v8f cU0 = {}, cU1 = {}, cV0 = {}, cV1 = {};
#pragma unroll 4
  for (int kk = 0; kk < Dd; kk += 4) {
    float2 av = *(const float2*)(aptr + kk);
    v2f a; a.x = av.x; a.y = av.y;
    float2 t; v2f b;
    t = *(const float2*)(b13a + kk); b.x = t.x; b.y = t.y; cU0 = wmma_f32(a, b, cU0);
    t = *(const float2*)(b13b + kk); b.x = t.x; b.y = t.y; cU1 = wmma_f32(a, b, cU1);
    t = *(const float2*)(b23a + kk); b.x = t.x; b.y = t.y; cV0 = wmma_f32(a, b, cV0);
    t = *(const float2*)(b23b + kk); b.x = t.x; b.y = t.y; cV1 = wmma_f32(a, b, cV1);
  }

  const int dA = d0 + n0 + lmod;
#pragma unroll
  for (int r = 0; r < 8; ++r) {
    int row = r0 + m0 + r + 8 * lhi;
    U[row * Dd + dA]      = cU0[r];
    U[row * Dd + dA + 16] = cU1[r];
    V[row * Dd + dA]      = cV0[r];
    V[row * Dd + dA + 16] = cV1[r];
  }
}

// ---------------------------------------------------------------------------
// Kernel 2: per-pair GEMM P = (hi*hj) @ W4^T, fused epilogue
//           out = tanh(U[i] + V[j] + P + bias), non-temporal stores.
// Grid: (129 pair-tiles, 4 d-tiles, 8 batches), block 256 (8 wave32).
// LDS: full h[b] (128x260) + W4 slice (64x260) + pair index tables ~= 196 KB.
// ---------------------------------------------------------------------------
__global__ __launch_bounds__(256, 1)
void pair_gemm_kernel(const float* __restrict__ h, const float* __restrict__ W,
                      const float* __restrict__ bias,
                      const float* __restrict__ U, const float* __restrict__ V,
                      float* __restrict__ out) {
  extern __shared__ float smem[];
  float* lds_h = smem;                        // 128 x STRH
  float* lds_w = lds_h + Nn * STRH;           // 64  x STRW : W4[d][k]
  int* lds_ii  = (int*)(lds_w + TDd * STRW);  // 64
  int* lds_jj  = lds_ii + TSp;                // 64

  const int b   = blockIdx.z;
  const int s0  = blockIdx.x * TSp;           // 129*64 == 8256, no ragged tile
  const int d0  = blockIdx.y * TDd;
  const int tid = threadIdx.x;

  // stage h[b] into LDS (padded rows)
  const float4* hg = (const float4*)(h + (size_t)b * Nn * Dd);
  for (int idx = tid; idx < Nn * (Dd / 4); idx += 256) {
    int row = idx >> 6, c4 = idx & 63;
    float4 v = hg[row * (Dd / 4) + c4];
    *(float4*)&lds_h[row * STRH + c4 * 4] = v;
  }
  // stage W4 slice [d0..d0+63][0..255] (float offset 768 within each W row)
  const float4* wg = (const float4*)W;
  for (int idx = tid; idx < TDd * (Dd / 4); idx += 256) {
    int row = idx >> 6, c4 = idx & 63;
    float4 v = wg[(size_t)(d0 + row) * 256 + 192 + c4];
    *(float4*)&lds_w[row * STRW + c4 * 4] = v;
  }
  // pair indices for this tile: closed-form + integer fixup
  if (tid < TSp) {
    int s = s0 + tid;
    float disc = sqrtf((float)(66049 - 8 * s));     // (2N+1)^2 - 8s, N=128
    int i = (int)((257.0f - disc) * 0.5f);
    i = i < 0 ? 0 : (i > Nn - 1 ? Nn - 1 : i);
    while (i < Nn - 1 && rowStart(i + 1) <= s) ++i;
    while (i > 0 && rowStart(i) > s) --i;
    lds_ii[tid] = i;
    lds_jj[tid] = i + (s - rowStart(i));
  }
  __syncthreads();

  const int l = tid & 31, wv = tid >> 5;
  const int lmod = l & 15, lhi = l >> 4;
  const int m0 = (wv & 3) * 16, n0 = (wv >> 2) * 32;

  // A fragment rows: this wave's 16 pairs; lane supplies K = 2*lhi + {0,1}
  const int ip = lds_ii[m0 + lmod];
  const int jp = lds_jj[m0 + lmod];
  const float* hi_ptr = lds_h + ip * STRH + 2 * lhi;
  const float* hj_ptr = lds_h + jp * STRH + 2 * lhi;
  const float* b0_ptr = lds_w + (n0 + lmod) * STRW + 2 * lhi;
  const float* b1_ptr = b0_ptr + 16 * STRW;

  v8f c0 = {}, c1 = {};
#pragma unroll 8
  for (int kk = 0; kk < Dd; kk += 4) {
    float2 hiv = *(const float2*)(hi_ptr + kk);
    float2 hjv = *(const float2*)(hj_ptr + kk);
    v2f a; a.x = hiv.x * hjv.x; a.y = hiv.y * hjv.y;   // on-the-fly hi*hj
    float2 w0 = *(const float2*)(b0_ptr + kk);
    float2 w1 = *(const float2*)(b1_ptr + kk);
    v2f bb0; bb0.x = w0.x; bb0.y = w0.y;
    v2f bb1; bb1.x = w1.x; bb1.y = w1.y;
    c0 = wmma_f32(a, bb0, c0);
    c1 = wmma_f32(a, bb1, c1);
  }

  // epilogue: out = tanh(U[i] + V[j] + P + bias)
  const float* Ub = U + (size_t)b * Nn * Dd;
  const float* Vb = V + (size_t)b * Nn * Dd;
  const int dA = d0 + n0 + lmod;
  const int dB = dA + 16;
  const float biasA = bias[dA];
  const float biasB = bias[dB];
  const int sBase = b * Sp + s0;
#pragma unroll
  for (int r = 0; r < 8; ++r) {
    int p = m0 + r + 8 * lhi;
    int i2 = lds_ii[p], j2 = lds_jj[p];
    float sum0 = Ub[i2 * Dd + dA] + Vb[j2 * Dd + dA] + biasA + c0[r];
    float sum1 = Ub[i2 * Dd + dB] + Vb[j2 * Dd + dB] + biasB + c1[r];
    size_t o = (size_t)(sBase + p) * Dd;
    __builtin_nontemporal_store(fast_tanh(sum0), &out[o + dA]);
    __builtin_nontemporal_store(fast_tanh(sum1), &out[o + dB]);
  }
}

// ---------------------------------------------------------------------------
extern "C" void kernel_launch(void* const* d_in, const int* in_sizes, int n_in,
                              void* d_out, int out_size, void* d_ws, size_t ws_size,
                              hipStream_t stream) {
  const float* h    = (const float*)d_in[0];   // (8,128,256) f32
  const float* W    = (const float*)d_in[1];   // (256,1024)  f32
  const float* bias = (const float*)d_in[2];   // (256,)      f32
  float* out = (float*)d_out;                  // (8,8256,256) f32

  float* U = (float*)d_ws;                     // (1024,256) f32 = 1 MB
  float* V = U + (size_t)8 * Nn * Dd;          // (1024,256) f32 = 1 MB

  const size_t smemUV   = (size_t)(64 * STRH + 2 * 64 * STRW) * sizeof(float);
  const size_t smemMain = (size_t)(Nn * STRH + TDd * STRW) * sizeof(float)
                        + 2 * TSp * sizeof(int);

  // allow >64KB dynamic LDS (WGP has 320KB)
  (void)hipFuncSetAttribute(reinterpret_cast<const void*>(uv_precompute_kernel),
                            hipFuncAttributeMaxDynamicSharedMemorySize, (int)smemUV);
  (void)hipFuncSetAttribute(reinterpret_cast<const void*>(pair_gemm_kernel),
                            hipFuncAttributeMaxDynamicSharedMemorySize, (int)smemMain);

  uv_precompute_kernel<<<dim3(16, 4), 256, smemUV, stream>>>(h, W, U, V);
  pair_gemm_kernel<<<dim3(129, 4, 8), 256, smemMain, stream>>>(h, W, bias, U, V, out);
}